// Model_21775484190778
// MI455X (gfx1250) — compile-verified
//
#include <hip/hip_runtime.h>
#include <math.h>

// ---------------------------------------------------------------------------
// Problem constants (from the reference)
// ---------------------------------------------------------------------------
constexpr int   NG   = 1024;          // gaussians (32*32)
constexpr int   KK   = 25;            // 5x5 patch
constexpr int   S    = NG * KK;       // 25600 samples
constexpr float DXC  = 2.0f / 31.0f;  // grid spacing
constexpr int   LDH  = 80;            // padded feature width (75->80, also h width)
constexpr int   NCO  = 64;            // padded conv output width (50->64)

typedef __attribute__((ext_vector_type(2))) float v2f;
typedef __attribute__((ext_vector_type(8))) float v8f;

// ---------------------------------------------------------------------------
// Workspace layout (floats)
// ---------------------------------------------------------------------------
constexpr size_t OFS_GAUSS = 0;                         // NG*8       packed per-gaussian
constexpr size_t OFS_IMG   = OFS_GAUSS + NG * 8;        // NG*80      img features (padded)
constexpr size_t OFS_WC    = OFS_IMG   + NG * LDH;      // 4*80*64    repacked conv weights
constexpr size_t OFS_H     = OFS_WC    + 4 * LDH * NCO; // 4*NG*80    hidden (conv||emb)
constexpr size_t OFS_H2    = OFS_H     + 4 * NG * LDH;  // 4*NG*80    hidden after lin1

// ---------------------------------------------------------------------------
// 1) Per-gaussian precompute: conic = inv(L L^T), pack [mx,my,c00,c01,c11,u,trC,0]
// ---------------------------------------------------------------------------
__global__ void prep_gauss(const float* __restrict__ means, const float* __restrict__ u,
                           const float* __restrict__ scaling, const float* __restrict__ transform,
                           float* __restrict__ g) {
    int n = blockIdx.x * blockDim.x + threadIdx.x;
    if (n >= NG) return;
    float s0 = scaling[n * 2 + 0], s1 = scaling[n * 2 + 1], t = transform[n];
    // cov = L L^T, L = [[s0,0],[t,s1]] -> [[s0^2, s0*t],[s0*t, t^2+s1^2]]
    float a = s0 * s0, b = s0 * t, d = t * t + s1 * s1;
    float inv = 1.0f / (a * d - b * b);            // det = (s0*s1)^2
    float c00 = d * inv, c01 = -b * inv, c11 = a * inv;
    g[n * 8 + 0] = means[n * 2 + 0];
    g[n * 8 + 1] = means[n * 2 + 1];
    g[n * 8 + 2] = c00;
    g[n * 8 + 3] = c01;
    g[n * 8 + 4] = c11;
    g[n * 8 + 5] = u[n];
    g[n * 8 + 6] = c00 + c11;
    g[n * 8 + 7] = 0.0f;
}

// ---------------------------------------------------------------------------
// 2) Repack conv_w (4,50,3,5,5) -> WcP[m][p][o], p=c*25+hw padded to 80, o padded to 64
// ---------------------------------------------------------------------------
__global__ void repack_conv_w(const float* __restrict__ conv_w, float* __restrict__ WcP) {
    int id = blockIdx.x * blockDim.x + threadIdx.x;
    if (id >= 4 * LDH * NCO) return;
    int m = id / (LDH * NCO);
    int r = id % (LDH * NCO);
    int p = r / NCO, o = r % NCO;
    float v = 0.0f;
    if (p < 75 && o < 50) v = conv_w[(m * 50 + o) * 75 + p];   // (3,5,5) flattens to p
    WcP[id] = v;
}

// ---------------------------------------------------------------------------
// 3) Mixture evaluation: per sample, loop all gaussians (table staged in LDS).
//    Writes padded img matrix: row i = [u_s(25) | pde(25) | mask(25) | 0(5)]
// ---------------------------------------------------------------------------
__global__ __launch_bounds__(256) void mixture(const float* __restrict__ means,
                                               const float* __restrict__ gauss,
                                               float* __restrict__ imgF) {
    __shared__ float4 gs[NG * 2];                          // 32 KB of 320 KB LDS
    for (int k = threadIdx.x; k < NG * 2; k += 256)
        gs[k] = ((const float4*)gauss)[k];
    __syncthreads();

    int s = blockIdx.x * 256 + threadIdx.x;                // 100 blocks * 256 == S exactly
    int i  = s / KK, hw = s % KK;
    int hh = hw / 5, ww = hw % 5;
    float sx = means[i * 2 + 0] + (float)(hh - 2) * DXC;
    float sy = means[i * 2 + 1] + (float)(ww - 2) * DXC;

    float acc_u = 0.0f, acc_p = 0.0f;
    for (int n = 0; n < NG; ++n) {
        float4 g0 = gs[n * 2 + 0];
        float4 g1 = gs[n * 2 + 1];
        float dx = sx - g0.x, dy = sy - g0.y;
        float cd0 = g0.z * dx + g0.w * dy;                 // (C d)_0
        float cd1 = g0.w * dx + g1.x * dy;                 // (C d)_1
        float q   = dx * cd0 + dy * cd1;
        float w   = __expf(-0.5f * q) * g1.y;              // exp * u_n
        acc_u += w;
        acc_p += w * (cd0 * cd0 + cd1 * cd1 - g1.z);       // trace of Hessian contrib
    }
    float mask = (fabsf(sx) < 1.0f && fabsf(sy) < 1.0f) ? 1.0f : 0.0f;

    float* row = imgF + (size_t)i * LDH;
    row[hw]       = acc_u;
    row[25 + hw]  = acc_p;
    row[50 + hw]  = mask;
    if (hw < 5) row[75 + hw] = 0.0f;                       // K-padding
}

// ---------------------------------------------------------------------------
// 4) Embedding branch (K=6, scalar): H[m][i][50+q] = tanh(params . emb_w + b)
// ---------------------------------------------------------------------------
__global__ void emb_kernel(const float* __restrict__ means, const float* __restrict__ u,
                           const float* __restrict__ scaling, const float* __restrict__ transform,
                           const float* __restrict__ emb_w, const float* __restrict__ emb_b,
                           float* __restrict__ H) {
    int id = blockIdx.x * blockDim.x + threadIdx.x;
    if (id >= 4 * NG * 30) return;
    int m = id / (NG * 30);
    int r = id % (NG * 30);
    int i = r / 30, q = r % 30;
    const float* W = emb_w + (m * 6) * 30 + q;
    float acc = emb_b[m * 30 + q];
    acc += means[i * 2 + 0] * W[0]   + means[i * 2 + 1] * W[30] + u[i] * W[60];
    acc += scaling[i * 2 + 0] * W[90] + scaling[i * 2 + 1] * W[120] + transform[i] * W[150];
    H[(size_t)m * NG * LDH + (size_t)i * LDH + 50 + q] = tanhf(acc);
}

// ---------------------------------------------------------------------------
// WMMA f32 16x16 tile: acc += A(16xK) @ B(Kx16), K multiple of 4, one wave.
// A layout: lane=row M; a.{x,y}=K {0,1} (lanes 0-15) / {2,3} (lanes 16-31).
// B layout: lane=col N; b.{x,y}=K rows {0,1} / {2,3} per half-wave.
// ---------------------------------------------------------------------------
template <int KDIM>
__device__ inline v8f wmma_tile_f32(const float* __restrict__ A, int lda,
                                    const float* __restrict__ B, int ldb) {
    int lane = threadIdx.x & 31;
    int half = lane >> 4;            // 0: lanes 0-15, 1: lanes 16-31
    int lr   = lane & 15;
    v8f acc = {0.f, 0.f, 0.f, 0.f, 0.f, 0.f, 0.f, 0.f};
#pragma unroll
    for (int k = 0; k < KDIM; k += 4) {
        v2f a, b;
        const float* Ar = A + lr * lda + k + half * 2;
        a.x = Ar[0];
        a.y = Ar[1];
        const float* Bc = B + (k + half * 2) * ldb + lr;
        b.x = Bc[0];
        b.y = Bc[ldb];
        acc = __builtin_amdgcn_wmma_f32_16x16x4_f32(false, a, false, b,
                                                    (short)0, acc, false, false);
    }
    return acc;
}

// ---------------------------------------------------------------------------
// 5) Conv-as-GEMM: Y[m] = tanh(imgF(1024x80) @ WcP[m](80x64) + conv_b) -> H cols 0..49
//    grid = 4 nets * 64 Mtiles * 4 Ntiles = 1024 waves
// ---------------------------------------------------------------------------
__global__ __launch_bounds__(32) void conv_gemm(const float* __restrict__ imgF,
                                                const float* __restrict__ WcP,
                                                const float* __restrict__ conv_b,
                                                float* __restrict__ H) {
    int id  = blockIdx.x;
    int m   = id >> 8;
    int rem = id & 255;
    int i0  = (rem >> 2) * 16, j0 = (rem & 3) * 16;
    v8f acc = wmma_tile_f32<LDH>(imgF + (size_t)i0 * LDH, LDH,
                                 WcP + (size_t)m * LDH * NCO + j0, NCO);
    int lane = threadIdx.x & 31;
    int half = lane >> 4, lr = lane & 15;
    int col = j0 + lr;
    if (col < 50) {
        float bias = conv_b[m * 50 + col];
        float* out = H + (size_t)m * NG * LDH + (size_t)(i0 + half * 8) * LDH + col;
#pragma unroll
        for (int v = 0; v < 8; ++v) out[v * LDH] = tanhf(acc[v] + bias);
    }
}

// ---------------------------------------------------------------------------
// 6) lin1: H2[m] = tanh(H(1024x80) @ lin1_w[m](80x80) + lin1_b)
//    grid = 4 * 64 * 5 = 1280 waves
// ---------------------------------------------------------------------------
__global__ __launch_bounds__(32) void lin1_gemm(const float* __restrict__ H,
                                                const float* __restrict__ lin1_w,
                                                const float* __restrict__ lin1_b,
                                                float* __restrict__ H2) {
    int id  = blockIdx.x;
    int m   = id / 320;
    int rem = id % 320;
    int i0  = (rem / 5) * 16, j0 = (rem % 5) * 16;
    v8f acc = wmma_tile_f32<LDH>(H + (size_t)m * NG * LDH + (size_t)i0 * LDH, LDH,
                                 lin1_w + (size_t)m * LDH * LDH + j0, LDH);
    int lane = threadIdx.x & 31;
    int half = lane >> 4, lr = lane & 15;
    int col = j0 + lr;
    float bias = lin1_b[m * LDH + col];
    float* out = H2 + (size_t)m * NG * LDH + (size_t)(i0 + half * 8) * LDH + col;
#pragma unroll
    for (int v = 0; v < 8; ++v) out[v * LDH] = tanhf(acc[v] + bias);
}

// ---------------------------------------------------------------------------
// 7) Output heads
// ---------------------------------------------------------------------------
__global__ void heads(const float* __restrict__ H2, const float* __restrict__ u,
                      const float* __restrict__ means, const float* __restrict__ scaling,
                      const float* __restrict__ transform,
                      const float* __restrict__ sol_w, const float* __restrict__ sol_b,
                      const float* __restrict__ tr_w,  const float* __restrict__ tr_b,
                      const float* __restrict__ sc_w,  const float* __restrict__ sc_b,
                      const float* __restrict__ tf_w,  const float* __restrict__ tf_b,
                      float* __restrict__ out) {
    int i = blockIdx.x * blockDim.x + threadIdx.x;
    if (i >= NG) return;
    const float* h0 = H2 + 0 * (size_t)NG * LDH + (size_t)i * LDH;
    const float* h1 = H2 + 1 * (size_t)NG * LDH + (size_t)i * LDH;
    const float* h2 = H2 + 2 * (size_t)NG * LDH + (size_t)i * LDH;
    const float* h3 = H2 + 3 * (size_t)NG * LDH + (size_t)i * LDH;
    float du = 0.f, dm0 = 0.f, dm1 = 0.f, ds0 = 0.f, ds1 = 0.f, dt = 0.f;
#pragma unroll 4
    for (int k = 0; k < LDH; ++k) {
        du  += h0[k] * sol_w[k];
        dm0 += h1[k] * tr_w[k * 2 + 0];
        dm1 += h1[k] * tr_w[k * 2 + 1];
        ds0 += h2[k] * sc_w[k * 2 + 0];
        ds1 += h2[k] * sc_w[k * 2 + 1];
        dt  += h3[k] * tf_w[k];
    }
    out[i * 6 + 0] = u[i] + du + sol_b[0];
    out[i * 6 + 1] = means[i * 2 + 0] + dm0 + tr_b[0];
    out[i * 6 + 2] = means[i * 2 + 1] + dm1 + tr_b[1];
    out[i * 6 + 3] = scaling[i * 2 + 0] * expf(ds0 + sc_b[0]);
    out[i * 6 + 4] = scaling[i * 2 + 1] * expf(ds1 + sc_b[1]);
    out[i * 6 + 5] = transform[i] + dt + tf_b[0];
}

// ---------------------------------------------------------------------------
// Launcher
// ---------------------------------------------------------------------------
extern "C" void kernel_launch(void* const* d_in, const int* in_sizes, int n_in,
                              void* d_out, int out_size, void* d_ws, size_t ws_size,
                              hipStream_t stream) {
    (void)in_sizes; (void)n_in; (void)out_size; (void)ws_size;
    const float* means     = (const float*)d_in[0];
    const float* u         = (const float*)d_in[1];
    const float* scaling   = (const float*)d_in[2];
    const float* transform = (const float*)d_in[3];
    const float* conv_w    = (const float*)d_in[4];
    const float* conv_b    = (const float*)d_in[5];
    const float* emb_w     = (const float*)d_in[6];
    const float* emb_b     = (const float*)d_in[7];
    const float* lin1_w    = (const float*)d_in[8];
    const float* lin1_b    = (const float*)d_in[9];
    const float* sol_w     = (const float*)d_in[10];
    const float* sol_b     = (const float*)d_in[11];
    const float* tr_w      = (const float*)d_in[12];
    const float* tr_b      = (const float*)d_in[13];
    const float* sc_w      = (const float*)d_in[14];
    const float* sc_b      = (const float*)d_in[15];
    const float* tf_w      = (const float*)d_in[16];
    const float* tf_b      = (const float*)d_in[17];
    float* out = (float*)d_out;
    float* ws  = (float*)d_ws;

    float* gauss = ws + OFS_GAUSS;
    float* imgF  = ws + OFS_IMG;
    float* WcP   = ws + OFS_WC;
    float* H     = ws + OFS_H;
    float* H2    = ws + OFS_H2;

    prep_gauss<<<(NG + 255) / 256, 256, 0, stream>>>(means, u, scaling, transform, gauss);
    repack_conv_w<<<(4 * LDH * NCO + 255) / 256, 256, 0, stream>>>(conv_w, WcP);
    mixture<<<S / 256, 256, 0, stream>>>(means, gauss, imgF);
    emb_kernel<<<(4 * NG * 30 + 255) / 256, 256, 0, stream>>>(means, u, scaling, transform,
                                                              emb_w, emb_b, H);
    conv_gemm<<<1024, 32, 0, stream>>>(imgF, WcP, conv_b, H);
    lin1_gemm<<<1280, 32, 0, stream>>>(H, lin1_w, lin1_b, H2);
    heads<<<(NG + 255) / 256, 256, 0, stream>>>(H2, u, means, scaling, transform,
                                                sol_w, sol_b, tr_w, tr_b,
                                                sc_w, sc_b, tf_w, tf_b, out);
}